// QuantumEEGNet_23278722744720
// MI455X (gfx1250) — compile-verified
//
#include <hip/hip_runtime.h>
#include <math.h>

// CDNA5 fp32 WMMA fragment types: 16x16x4 -> A,B are 2 VGPRs (64 f32 / 32 lanes), C/D 8 VGPRs.
typedef float v2f __attribute__((ext_vector_type(2)));
typedef float v8f __attribute__((ext_vector_type(8)));

#define BN_EPS 1e-5f

__device__ __forceinline__ float elu1(float x) {
    return x > 0.0f ? x : (__expf(x) - 1.0f);
}

__launch_bounds__(128)
__global__ void qeegnet_fused(
    const float* __restrict__ x,    // (B,1,2,127)
    const float* __restrict__ w1,   // (8,1,1,64)
    const float* __restrict__ g1, const float* __restrict__ b1,
    const float* __restrict__ m1, const float* __restrict__ v1,
    const float* __restrict__ w2,   // (16,1,2,1)
    const float* __restrict__ g2, const float* __restrict__ b2,
    const float* __restrict__ m2, const float* __restrict__ v2,
    const float* __restrict__ w3,   // (16,16,1,16)
    const float* __restrict__ g3, const float* __restrict__ b3,
    const float* __restrict__ m3, const float* __restrict__ v3,
    const float* __restrict__ w4,   // (16,16,1,1)
    const float* __restrict__ g4, const float* __restrict__ b4,
    const float* __restrict__ m4, const float* __restrict__ v4,
    const float* __restrict__ qw,   // (2,4)
    const float* __restrict__ fcw,  // (2,64)
    const float* __restrict__ fcb,  // (2,)
    float* __restrict__ out)        // (B,2)
{
    // ---- LDS (~42 KB per workgroup) ----
    __shared__ float xpad[2 * 192];     // x padded by 32 each side (zero)
    __shared__ float w1s[16 * 64];      // conv1 weights zero-padded to 16 rows (WMMA A)
    __shared__ float w3s[16 * 256];     // [co][ci*16+k]
    __shared__ float w4s[16 * 16];      // [co][ci]
    __shared__ float c1s[8 * 256];      // conv1+bn+elu: [f][h*128+t]
    __shared__ float p2pad[16 * 64];    // pooled (16x32) padded by 8 (zero), offset +8
    __shared__ float c3s[16 * 48];      // conv3+bn+elu: [co][p], p=0..47 (valid 0..32)
    __shared__ float c4s[16 * 48];      // conv4+bn+elu
    __shared__ float pooled[64];        // (16 ch x 4) quantum angles
    __shared__ float qouts[64];         // (16 ch x 4) Z expectations
    __shared__ float sc1[8], bi1[8];
    __shared__ float sc2[16], bi2[16], sc3[16], bi3[16], sc4[16], bi4[16];

    const int tid   = threadIdx.x;
    const int bidx  = blockIdx.x;
    const int lane  = tid & 31;
    const int wid   = tid >> 5;   // wave id 0..3 (wave32)
    const int laneM = lane & 15;  // matrix row / col within halves
    const int hiK   = lane >> 4;  // lane half selects K pair {0,1} vs {2,3}

    // ---- stage 0: preload weights, zero-init padded images, BN fold ----
    for (int i = tid; i < 2 * 192; i += 128) xpad[i]  = 0.0f;
    for (int i = tid; i < 16 * 64; i += 128) p2pad[i] = 0.0f;
    for (int i = tid; i < 1024; i += 128) w1s[i] = (i < 512) ? w1[i] : 0.0f;  // rows 8..15 zero
    for (int i = tid; i < 4096; i += 128) w3s[i] = w3[i];
    for (int i = tid; i < 256;  i += 128) w4s[i] = w4[i];
    if (tid < 16) {
        float s;
        s = g2[tid] * rsqrtf(v2[tid] + BN_EPS); sc2[tid] = s; bi2[tid] = b2[tid] - m2[tid] * s;
        s = g3[tid] * rsqrtf(v3[tid] + BN_EPS); sc3[tid] = s; bi3[tid] = b3[tid] - m3[tid] * s;
        s = g4[tid] * rsqrtf(v4[tid] + BN_EPS); sc4[tid] = s; bi4[tid] = b4[tid] - m4[tid] * s;
        if (tid < 8) {
            s = g1[tid] * rsqrtf(v1[tid] + BN_EPS); sc1[tid] = s; bi1[tid] = b1[tid] - m1[tid] * s;
        }
    }
    for (int i = tid; i < 254; i += 128) {
        int h = i / 127, t = i - h * 127;
        xpad[h * 192 + 32 + t] = x[(size_t)bidx * 254 + i];
    }
    __syncthreads();

    // ---- stage 1: conv1 (+bn1+elu) as WMMA GEMM: [16pad x 64] x [64 x 256] ----
    // N index n = h*128 + t ; all 4 waves, 16 N-tiles.
    for (int tt = wid; tt < 16; tt += 4) {
        v8f acc = {};
        const int n = tt * 16 + laneM;
        const int h = n >> 7, t = n & 127;
        for (int s = 0; s < 16; ++s) {
            const int k0 = 4 * s + 2 * hiK;
            v2f a, bfrag;
            a.x = w1s[laneM * 64 + k0];       // rows 8..15 are zero-padded data
            a.y = w1s[laneM * 64 + k0 + 1];
            bfrag.x = xpad[h * 192 + t + k0];
            bfrag.y = xpad[h * 192 + t + k0 + 1];
            acc = __builtin_amdgcn_wmma_f32_16x16x4_f32(
                false, a, false, bfrag, (short)0, acc, false, false);
        }
        if (hiK == 0) {  // only rows 0..7 (real channels); acc[v] = out[f=v][n]
            #pragma unroll
            for (int v = 0; v < 8; ++v)
                c1s[v * 256 + n] = elu1(acc[v] * sc1[v] + bi1[v]);
        }
    }
    __syncthreads();

    // ---- stage 2: grouped conv2 (+bn2+elu) + pool4 -> p2pad[ch][8+q], q=0..31 ----
    for (int i = tid; i < 512; i += 128) {
        const int ch = i >> 5, q = i & 31;
        const int g  = ch >> 1;
        const float wa = w2[ch * 2 + 0], wb = w2[ch * 2 + 1];
        float s = 0.0f;
        #pragma unroll
        for (int r = 0; r < 4; ++r) {
            const int t = q * 4 + r;
            const float c = wa * c1s[g * 256 + t] + wb * c1s[g * 256 + 128 + t];
            s += elu1(c * sc2[ch] + bi2[ch]);
        }
        p2pad[ch * 64 + 8 + q] = 0.25f * s;
    }
    __syncthreads();

    // ---- stage 3: conv3 (+bn3+elu) as WMMA GEMM: [16 x 256] x [256 x 48] ----
    // B[k=ci*16+kt][p] = p2pad[ci][(p-8+kt)+8] = p2pad[ci*64 + p + kt]; waves 0..2.
    if (wid < 3) {
        const int p = wid * 16 + laneM;
        v8f acc = {};
        for (int s = 0; s < 64; ++s) {
            const int k0 = 4 * s + 2 * hiK;
            const int k1 = k0 + 1;
            v2f a, bfrag;
            a.x = w3s[laneM * 256 + k0];
            a.y = w3s[laneM * 256 + k1];
            bfrag.x = p2pad[(k0 >> 4) * 64 + p + (k0 & 15)];
            bfrag.y = p2pad[(k1 >> 4) * 64 + p + (k1 & 15)];
            acc = __builtin_amdgcn_wmma_f32_16x16x4_f32(
                false, a, false, bfrag, (short)0, acc, false, false);
        }
        #pragma unroll
        for (int v = 0; v < 8; ++v) {
            const int co = v + 8 * hiK;
            c3s[co * 48 + p] = elu1(acc[v] * sc3[co] + bi3[co]);
        }
    }
    __syncthreads();

    // ---- stage 4: conv4 pointwise (+bn4+elu) as WMMA GEMM: [16 x 16] x [16 x 48] ----
    if (wid < 3) {
        const int p = wid * 16 + laneM;
        v8f acc = {};
        #pragma unroll
        for (int s = 0; s < 4; ++s) {
            const int k0 = 4 * s + 2 * hiK;
            v2f a, bfrag;
            a.x = w4s[laneM * 16 + k0];
            a.y = w4s[laneM * 16 + k0 + 1];
            bfrag.x = c3s[k0 * 48 + p];
            bfrag.y = c3s[(k0 + 1) * 48 + p];
            acc = __builtin_amdgcn_wmma_f32_16x16x4_f32(
                false, a, false, bfrag, (short)0, acc, false, false);
        }
        #pragma unroll
        for (int v = 0; v < 8; ++v) {
            const int co = v + 8 * hiK;
            c4s[co * 48 + p] = elu1(acc[v] * sc4[co] + bi4[co]);
        }
    }
    __syncthreads();

    // ---- stage 5: pool8 over first 32 positions -> 16ch x 4 quantum angles ----
    if (tid < 64) {
        const int co = tid >> 2, q = tid & 3;
        float s = 0.0f;
        #pragma unroll
        for (int r = 0; r < 8; ++r) s += c4s[co * 48 + q * 8 + r];
        pooled[tid] = s * 0.125f;
    }
    __syncthreads();

    // ---- stage 6: 4-qubit real statevector sim, one thread per channel ----
    if (tid < 16) {
        float st[16];
        #pragma unroll
        for (int i = 0; i < 16; ++i) st[i] = 0.0f;
        st[0] = 1.0f;

        // encoding RY layer (per-item angles)
        #pragma unroll
        for (int w = 0; w < 4; ++w) {
            float sn, cs;
            __sincosf(0.5f * pooled[tid * 4 + w], &sn, &cs);
            const int stride = 1 << (3 - w);
            #pragma unroll
            for (int i0 = 0; i0 < 16; ++i0) {
                if (i0 & stride) continue;
                const int i1 = i0 | stride;
                const float a0 = st[i0], a1 = st[i1];
                st[i0] = cs * a0 - sn * a1;
                st[i1] = sn * a0 + cs * a1;
            }
        }
        // 2 variational layers: CNOT ring then RY(weights)
        for (int j = 0; j < 2; ++j) {
            #pragma unroll
            for (int k = 0; k < 4; ++k) {
                const int t  = (k + 1) & 3;
                const int tb = 1 << (3 - t);
                float tmp[16];
                #pragma unroll
                for (int i = 0; i < 16; ++i)
                    tmp[i] = st[((i >> (3 - k)) & 1) ? (i ^ tb) : i];
                #pragma unroll
                for (int i = 0; i < 16; ++i) st[i] = tmp[i];
            }
            #pragma unroll
            for (int w = 0; w < 4; ++w) {
                float sn, cs;
                __sincosf(0.5f * qw[j * 4 + w], &sn, &cs);
                const int stride = 1 << (3 - w);
                #pragma unroll
                for (int i0 = 0; i0 < 16; ++i0) {
                    if (i0 & stride) continue;
                    const int i1 = i0 | stride;
                    const float a0 = st[i0], a1 = st[i1];
                    st[i0] = cs * a0 - sn * a1;
                    st[i1] = sn * a0 + cs * a1;
                }
            }
        }
        // Z expectations
        #pragma unroll
        for (int w = 0; w < 4; ++w) {
            float z = 0.0f;
            #pragma unroll
            for (int i = 0; i < 16; ++i) {
                const float pr = st[i] * st[i];
                z += ((i >> (3 - w)) & 1) ? -pr : pr;
            }
            qouts[tid * 4 + w] = z;
        }
    }
    __syncthreads();

    // ---- stage 7: FC (2 x 64) ----
    if (tid < 2) {
        float acc = fcb[tid];
        #pragma unroll 8
        for (int i = 0; i < 64; ++i) acc += qouts[i] * fcw[tid * 64 + i];
        out[(size_t)bidx * 2 + tid] = acc;
    }
}

extern "C" void kernel_launch(void* const* d_in, const int* in_sizes, int n_in,
                              void* d_out, int out_size, void* d_ws, size_t ws_size,
                              hipStream_t stream) {
    (void)n_in; (void)d_ws; (void)ws_size; (void)out_size;
    const float* x   = (const float*)d_in[0];
    const float* w1  = (const float*)d_in[1];
    const float* g1  = (const float*)d_in[2];
    const float* b1  = (const float*)d_in[3];
    const float* m1  = (const float*)d_in[4];
    const float* v1  = (const float*)d_in[5];
    const float* w2  = (const float*)d_in[6];
    const float* g2  = (const float*)d_in[7];
    const float* b2  = (const float*)d_in[8];
    const float* m2  = (const float*)d_in[9];
    const float* v2  = (const float*)d_in[10];
    const float* w3  = (const float*)d_in[11];
    const float* g3  = (const float*)d_in[12];
    const float* b3  = (const float*)d_in[13];
    const float* m3  = (const float*)d_in[14];
    const float* v3  = (const float*)d_in[15];
    const float* w4  = (const float*)d_in[16];
    const float* g4  = (const float*)d_in[17];
    const float* b4  = (const float*)d_in[18];
    const float* m4  = (const float*)d_in[19];
    const float* v4  = (const float*)d_in[20];
    const float* qw  = (const float*)d_in[21];
    const float* fcw = (const float*)d_in[22];
    const float* fcb = (const float*)d_in[23];
    float* out = (float*)d_out;

    const int B = in_sizes[0] / (2 * 127);  // (B,1,2,127)
    qeegnet_fused<<<B, 128, 0, stream>>>(x, w1, g1, b1, m1, v1,
                                         w2, g2, b2, m2, v2,
                                         w3, g3, b3, m3, v3,
                                         w4, g4, b4, m4, v4,
                                         qw, fcw, fcb, out);
}